// RPN_69552700391911
// MI455X (gfx1250) — compile-verified
//
#include <hip/hip_runtime.h>
#include <math.h>

typedef __attribute__((ext_vector_type(2))) float v2f;
typedef __attribute__((ext_vector_type(8))) float v8f;

#define HH 50
#define WW 50
#define NPIX 2500
#define CIN 512
#define CMID 512
#define NIMG 4
#define NANC 9
#define R_TOT (NPIX * NANC)        // 22500
#define PROP_BEFORE 6000
#define PROP_AFTER 300
#define SORT_N 32768
#define NMS_WORDS 188              // ceil(6000/32)
#define NMS_TH 0.7f
#define MIN_SZ 16.0f

#define STRIP 32                   // pixels per block
#define HALO 51                    // |ky*50+kx| <= 51
#define TILE_PX (STRIP + 2 * HALO) // 134 staged pixels
#define NOCT (CMID / 16)           // 32 oc tiles
// apack: [oct][kpos][icq(128)][lane(32)][2] floats; per (oct,kpos): 8192 floats
#define APACK_OKSTRIDE 8192

static __device__ __forceinline__ v8f wmma_f32(v2f a, v2f b, v8f c) {
    return __builtin_amdgcn_wmma_f32_16x16x4_f32(false, a, false, b,
                                                 (short)0, c, false, false);
}

// ---------------------------------------------------------------------------
// Repack w1 (oc, ic, 3, 3) into WMMA A-fragment order so the conv kernel can
// fetch each lane's (a.x, a.y) with a single coalesced b64 load.
// ---------------------------------------------------------------------------
__global__ __launch_bounds__(256) void repack_w_kernel(
    const float* __restrict__ w1, float* __restrict__ apack)
{
    int gid = blockIdx.x * blockDim.x + threadIdx.x;   // 32*9*128*32 threads
    int lane = gid & 31;
    int t = gid >> 5;
    int icq = t & 127; t >>= 7;
    int kpos = t % 9;
    int oct = t / 9;
    int oc = oct * 16 + (lane & 15);
    int ic = icq * 4 + (lane >> 4) * 2;
    size_t src = (size_t)oc * (CIN * 9) + (size_t)ic * 9 + kpos;
    apack[(size_t)gid * 2 + 0] = w1[src];
    apack[(size_t)gid * 2 + 1] = w1[src + 9];
}

// ---------------------------------------------------------------------------
// 3x3 conv (512->512) + bias + ReLU, implicit GEMM on f32 WMMA.
// Block = 256 thr (8 waves) owns a 32-pixel strip x all 512 oc.
// B strip (134 px x 512 ic = 274KB) staged in LDS via async-to-LDS copies and
// shared by all waves; zero padding applied as a branchless mask multiply.
// Each wave: 4 oc-tiles x 2 px-tiles = 8 WMMA accumulators.
// ---------------------------------------------------------------------------
__global__ __launch_bounds__(256) void conv3x3_wmma_kernel(
    const float* __restrict__ xin, const float* __restrict__ apack,
    const float* __restrict__ b1, float* __restrict__ inter)
{
    extern __shared__ float btile[];                  // [512][TILE_PX]
    const int tid = threadIdx.x;
    const int wave = tid >> 5;
    const int lane = tid & 31;
    const int half = lane >> 4;
    const int lid  = lane & 15;
    const int p0 = blockIdx.x * STRIP;
    const int n  = blockIdx.z;
    const float* __restrict__ xn = xin + (size_t)n * CIN * NPIX;

    // ---- async fill: btile[ic][i] = xn[ic*NPIX + clamp(p0-HALO+i)]
    const int base = p0 - HALO;
    for (int idx = tid; idx < CIN * TILE_PX; idx += 256) {  // 268 exact iters
        int ic = idx / TILE_PX;
        int i  = idx - ic * TILE_PX;
        int px = base + i;
        px = px < 0 ? 0 : (px >= NPIX ? NPIX - 1 : px);
        unsigned ldsoff = (unsigned)(size_t)&btile[idx];
        unsigned long long gaddr =
            (unsigned long long)(size_t)(xn + (size_t)ic * NPIX + px);
        asm volatile("global_load_async_to_lds_b32 %0, %1, off"
                     :: "v"(ldsoff), "v"(gaddr) : "memory");
    }
    asm volatile("s_wait_asynccnt 0x0" ::: "memory");
    __syncthreads();

    // ---- pixel / oc coordinates
    int p_s[2]; bool pv[2]; int y_s[2], x_s[2];
#pragma unroll
    for (int s = 0; s < 2; ++s) {
        p_s[s] = p0 + s * 16 + lid;
        pv[s]  = p_s[s] < NPIX;
        y_s[s] = p_s[s] / WW;
        x_s[s] = p_s[s] % WW;
    }
    const int oct0 = wave * 4;                        // 4 contiguous oc tiles

    v8f acc[8];
#pragma unroll
    for (int q = 0; q < 8; ++q) acc[q] = (v8f){};

    for (int kpos = 0; kpos < 9; ++kpos) {
        const int ky = kpos / 3 - 1, kx = kpos % 3 - 1;
        const int koff = ky * WW + kx;
        float mf[2]; int pix[2];
#pragma unroll
        for (int s = 0; s < 2; ++s) {
            int yy = y_s[s] + ky, xx = x_s[s] + kx;
            mf[s] = (pv[s] && (unsigned)yy < (unsigned)HH &&
                              (unsigned)xx < (unsigned)WW) ? 1.0f : 0.0f;
            pix[s] = s * 16 + lid + koff + HALO;      // 0..133, block-invariant
        }
        const size_t abase =
            ((size_t)(oct0 * 9 + kpos)) * APACK_OKSTRIDE + (size_t)lane * 2;
#pragma unroll 2
        for (int icq = 0; icq < 128; ++icq) {
            const int ic = icq * 4 + half * 2;        // K = kk + half*2 + {0,1}
            v2f b[2];
#pragma unroll
            for (int s = 0; s < 2; ++s) {
                b[s].x = btile[(size_t)ic * TILE_PX + pix[s]] * mf[s];
                b[s].y = btile[(size_t)(ic + 1) * TILE_PX + pix[s]] * mf[s];
            }
            v2f a[4];
#pragma unroll
            for (int j = 0; j < 4; ++j)
                a[j] = *(const v2f*)(apack + abase +
                        (size_t)j * 9 * APACK_OKSTRIDE + (size_t)icq * 64);
#pragma unroll
            for (int j = 0; j < 4; ++j)
#pragma unroll
                for (int s = 0; s < 2; ++s)
                    acc[j * 2 + s] = wmma_f32(a[j], b[s], acc[j * 2 + s]);
        }
    }

    // ---- epilogue: bias + ReLU
#pragma unroll
    for (int j = 0; j < 4; ++j) {
        const int oct = oct0 + j;
#pragma unroll
        for (int s = 0; s < 2; ++s) {
            if (pv[s]) {
                float* outp = inter + (size_t)n * CMID * NPIX + p_s[s];
#pragma unroll
                for (int v = 0; v < 8; ++v) {
                    const int o = oct * 16 + v + half * 8;
                    float val = acc[j * 2 + s][v] + b1[o];
                    outp[(size_t)o * NPIX] = val > 0.0f ? val : 0.0f;
                }
            }
        }
    }
}

// ---------------------------------------------------------------------------
// 1x1 conv head (score: 18ch, loc: 36ch) with f32 WMMA, branchless edges.
// mode 0: score[(n, p*9+a)*2 + c], ch = a*2+c ; mode 1: loc, ch = a*4+c
// ---------------------------------------------------------------------------
__global__ __launch_bounds__(256) void conv1x1_wmma_kernel(
    const float* __restrict__ inter, const float* __restrict__ w,
    const float* __restrict__ bias, float* __restrict__ out,
    const int outC, const int mode)
{
    const int wave = threadIdx.x >> 5;
    const int lane = threadIdx.x & 31;
    const int half = lane >> 4;
    const int lid  = lane & 15;
    const int NT = (NPIX + 15) / 16;                  // 157 pixel tiles
    const int tile = blockIdx.x * 8 + wave;           // wave-uniform
    if (tile >= NT) return;
    const int p  = tile * 16 + lid;
    const bool pv = p < NPIX;
    const int pc = pv ? p : (NPIX - 1);
    const float pvf = pv ? 1.0f : 0.0f;
    const int oc  = blockIdx.y * 16 + lid;
    const int ocr = oc < outC ? oc : 0;
    const int n   = blockIdx.z;

    const float* __restrict__ ap = w + (size_t)ocr * CMID;
    const float* __restrict__ bp = inter + (size_t)n * CMID * NPIX + pc;

    v8f acc = (v8f){};
#pragma unroll 4
    for (int ic0 = 0; ic0 < CMID; ic0 += 4) {
        const int ic = ic0 + half * 2;
        v2f a = *(const v2f*)(ap + ic);               // contiguous b64
        v2f b;
        b.x = bp[(size_t)ic * NPIX] * pvf;
        b.y = bp[(size_t)(ic + 1) * NPIX] * pvf;
        acc = wmma_f32(a, b, acc);
    }
    if (pv) {
#pragma unroll
        for (int v = 0; v < 8; ++v) {
            const int ch = blockIdx.y * 16 + v + half * 8;
            if (ch < outC) {
                float val = acc[v] + bias[ch];
                if (mode == 0) {
                    size_t r = (size_t)p * NANC + (ch >> 1);
                    out[((size_t)n * R_TOT + r) * 2 + (ch & 1)] = val;
                } else {
                    size_t r = (size_t)p * NANC + (ch >> 2);
                    out[((size_t)n * R_TOT + r) * 4 + (ch & 3)] = val;
                }
            }
        }
    }
}

// ---------------------------------------------------------------------------
// Anchors + box decode + clip + min-size mask + softmax-fg score.
// ---------------------------------------------------------------------------
__global__ void decode_kernel(const float* __restrict__ loc,
                              const float* __restrict__ score,
                              float* __restrict__ anchors_out,
                              float* __restrict__ boxes,
                              float* __restrict__ fg,
                              const int* __restrict__ img_h_p,
                              const int* __restrict__ img_w_p)
{
    int gid = blockIdx.x * blockDim.x + threadIdx.x;
    if (gid >= NIMG * R_TOT) return;
    int n = gid / R_TOT, r = gid % R_TOT;
    int p = r / NANC, a = r % NANC;
    int y = p / WW, x = p % WW;
    int ir = a / 3, is = a % 3;
    const float ratios[3] = {0.5f, 1.0f, 2.0f};
    const float scales[3] = {8.0f, 16.0f, 32.0f};
    float ah = 16.0f * sqrtf(ratios[ir]) * scales[is];
    float aw = 16.0f * sqrtf(1.0f / ratios[ir]) * scales[is];
    float acy = 16.0f * (float)y + 8.0f;
    float acx = 16.0f * (float)x + 8.0f;
    float ay1 = acy - 0.5f * ah, ax1 = acx - 0.5f * aw;
    float ay2 = acy + 0.5f * ah, ax2 = acx + 0.5f * aw;
    if (n == 0) {
        anchors_out[(size_t)r * 4 + 0] = ay1;
        anchors_out[(size_t)r * 4 + 1] = ax1;
        anchors_out[(size_t)r * 4 + 2] = ay2;
        anchors_out[(size_t)r * 4 + 3] = ax2;
    }
    const float* l = loc + ((size_t)n * R_TOT + r) * 4;
    float cy = l[0] * ah + acy;
    float cx = l[1] * aw + acx;
    float bh = expf(l[2]) * ah;
    float bw = expf(l[3]) * aw;
    float imh = (float)img_h_p[0], imw = (float)img_w_p[0];
    float y1 = fminf(fmaxf(cy - 0.5f * bh, 0.0f), imh);
    float x1 = fminf(fmaxf(cx - 0.5f * bw, 0.0f), imw);
    float y2 = fminf(fmaxf(cy + 0.5f * bh, 0.0f), imh);
    float x2 = fminf(fmaxf(cx + 0.5f * bw, 0.0f), imw);
    float* bo = boxes + ((size_t)n * R_TOT + r) * 4;
    bo[0] = y1; bo[1] = x1; bo[2] = y2; bo[3] = x2;
    const float* s = score + ((size_t)n * R_TOT + r) * 2;
    float f = 1.0f / (1.0f + expf(s[0] - s[1]));      // softmax fg prob
    bool valid = ((y2 - y1) >= MIN_SZ) && ((x2 - x1) >= MIN_SZ);
    fg[(size_t)n * R_TOT + r] = valid ? f : -INFINITY;
}

// ---------------------------------------------------------------------------
// Full bitonic sort of 32768 (score,idx) pairs in 256KB of CDNA5 LDS;
// one workgroup (1024 thr = 32 waves) per image. Emits top-6000 sorted.
// ---------------------------------------------------------------------------
__global__ __launch_bounds__(1024) void sort_topk_kernel(
    const float* __restrict__ fg, const float* __restrict__ boxes,
    float* __restrict__ topscore, int* __restrict__ topidx,
    float* __restrict__ topbox)
{
    extern __shared__ char smem[];
    float* key = (float*)smem;                        // 32768 f32
    int* idx = (int*)(smem + SORT_N * sizeof(float)); // 32768 i32
    const int n = blockIdx.x;
    const int tid = threadIdx.x;
    const float* f = fg + (size_t)n * R_TOT;
    for (int i = tid; i < SORT_N; i += 1024) {
        float sc = (i < R_TOT) ? f[i] : -INFINITY;
        key[i] = -sc;                                 // ascending => score desc
        idx[i] = (i < R_TOT) ? i : 0;
    }
    __syncthreads();
    for (int k = 2; k <= SORT_N; k <<= 1) {
        for (int j = k >> 1; j > 0; j >>= 1) {
            for (int i = tid; i < SORT_N; i += 1024) {
                int ixj = i ^ j;
                if (ixj > i) {
                    bool up = ((i & k) == 0);
                    float ki = key[i], kj = key[ixj];
                    if ((ki > kj) == up) {
                        key[i] = kj; key[ixj] = ki;
                        int t = idx[i]; idx[i] = idx[ixj]; idx[ixj] = t;
                    }
                }
            }
            __syncthreads();
        }
    }
    for (int i = tid; i < PROP_BEFORE; i += 1024) {
        int id = idx[i];
        topscore[(size_t)n * PROP_BEFORE + i] = -key[i];
        topidx[(size_t)n * PROP_BEFORE + i] = id;
        const float* b = boxes + ((size_t)n * R_TOT + id) * 4;
        float* tb = topbox + ((size_t)n * PROP_BEFORE + i) * 4;
        tb[0] = b[0]; tb[1] = b[1]; tb[2] = b[2]; tb[3] = b[3];
    }
}

// ---------------------------------------------------------------------------
// Parallel NMS suppression bitmask: bit (i, j) set iff j>i and IoU>0.7.
// ---------------------------------------------------------------------------
__global__ void nms_mask_kernel(const float* __restrict__ topbox,
                                unsigned* __restrict__ mask)
{
    int gid = blockIdx.x * blockDim.x + threadIdx.x;
    const int total = NIMG * PROP_BEFORE * NMS_WORDS;
    if (gid >= total) return;
    int wword = gid % NMS_WORDS;
    int rest = gid / NMS_WORDS;
    int i = rest % PROP_BEFORE;
    int n = rest / PROP_BEFORE;
    const float* bi = topbox + ((size_t)n * PROP_BEFORE + i) * 4;
    float iy1 = bi[0], ix1 = bi[1], iy2 = bi[2], ix2 = bi[3];
    float ia = (iy2 - iy1) * (ix2 - ix1);
    unsigned bits = 0u;
    int jbase = wword * 32;
    for (int bmark = 0; bmark < 32; ++bmark) {
        int j = jbase + bmark;
        if (j <= i || j >= PROP_BEFORE) continue;
        const float* bj = topbox + ((size_t)n * PROP_BEFORE + j) * 4;
        float ty = fmaxf(iy1, bj[0]);
        float tx = fmaxf(ix1, bj[1]);
        float by = fminf(iy2, bj[2]);
        float bx = fminf(ix2, bj[3]);
        float inter = fmaxf(by - ty, 0.0f) * fmaxf(bx - tx, 0.0f);
        float ja = (bj[2] - bj[0]) * (bj[3] - bj[1]);
        float iou = inter / (ia + ja - inter + 1e-9f);
        if (iou > NMS_TH) bits |= (1u << bmark);
    }
    mask[((size_t)n * PROP_BEFORE + i) * NMS_WORDS + wword] = bits;
}

// ---------------------------------------------------------------------------
// Sequential greedy scan over the sorted 6000 using the precomputed masks;
// scatter first 300 kept boxes (zero-padded), roi_indices = image id.
// ---------------------------------------------------------------------------
__global__ __launch_bounds__(256) void nms_scan_kernel(
    const float* __restrict__ topscore, const float* __restrict__ topbox,
    const unsigned* __restrict__ mask, float* __restrict__ rois,
    int* __restrict__ ridx)
{
    __shared__ unsigned remv[NMS_WORDS];
    __shared__ int alive;
    __shared__ int cnt;
    const int n = blockIdx.x;
    const int tid = threadIdx.x;
    for (int wv = tid; wv < NMS_WORDS; wv += 256) remv[wv] = 0u;
    for (int i = tid; i < PROP_AFTER * 4; i += 256)
        rois[(size_t)n * PROP_AFTER * 4 + i] = 0.0f;
    for (int i = tid; i < PROP_AFTER; i += 256)
        ridx[(size_t)n * PROP_AFTER + i] = n;
    if (tid == 0) cnt = 0;
    __syncthreads();
    for (int i = 0; i < PROP_BEFORE; ++i) {
        if (tid == 0) {
            float sc = topscore[(size_t)n * PROP_BEFORE + i];
            alive = (isfinite(sc) && !((remv[i >> 5] >> (i & 31)) & 1u)) ? 1 : 0;
        }
        __syncthreads();
        if (alive) {
            if (tid < NMS_WORDS)
                remv[tid] |= mask[((size_t)n * PROP_BEFORE + i) * NMS_WORDS + tid];
            if (tid == 0) {
                if (cnt < PROP_AFTER) {
                    const float* b = topbox + ((size_t)n * PROP_BEFORE + i) * 4;
                    float* o = rois + ((size_t)n * PROP_AFTER + cnt) * 4;
                    o[0] = b[0]; o[1] = b[1]; o[2] = b[2]; o[3] = b[3];
                }
                cnt = cnt + 1;
            }
        }
        __syncthreads();
    }
}

// ---------------------------------------------------------------------------
extern "C" void kernel_launch(void* const* d_in, const int* in_sizes, int n_in,
                              void* d_out, int out_size, void* d_ws, size_t ws_size,
                              hipStream_t stream)
{
    (void)in_sizes; (void)n_in; (void)out_size; (void)ws_size;
    const float* x       = (const float*)d_in[0];
    const float* w1      = (const float*)d_in[1];
    const float* b1      = (const float*)d_in[2];
    const float* w_score = (const float*)d_in[3];
    const float* b_score = (const float*)d_in[4];
    const float* w_loc   = (const float*)d_in[5];
    const float* b_loc   = (const float*)d_in[6];
    const int* img_h     = (const int*)d_in[7];
    const int* img_w     = (const int*)d_in[8];

    // d_out layout: loc(4*22500*4) | score(4*22500*2) | rois(4*300*4) |
    //               roi_indices(4*300 int32) | anchor(22500*4)
    float* out = (float*)d_out;
    float* out_loc    = out;
    float* out_score  = out_loc + (size_t)NIMG * R_TOT * 4;
    float* out_rois   = out_score + (size_t)NIMG * R_TOT * 2;
    int*   out_ridx   = (int*)(out_rois + (size_t)NIMG * PROP_AFTER * 4);
    float* out_anchor = (float*)(out_ridx + (size_t)NIMG * PROP_AFTER);

    // workspace carve-out (256B aligned), ~51 MB total
    char* wsb = (char*)d_ws;
    size_t off = 0;
    auto carve = [&](size_t bytes) -> void* {
        void* p = wsb + off;
        off += (bytes + 255) & ~(size_t)255;
        return p;
    };
    float*    inter    = (float*)carve((size_t)NIMG * CMID * NPIX * 4);
    float*    apack    = (float*)carve((size_t)NOCT * 9 * APACK_OKSTRIDE * 4);
    float*    boxes    = (float*)carve((size_t)NIMG * R_TOT * 4 * 4);
    float*    fg       = (float*)carve((size_t)NIMG * R_TOT * 4);
    float*    topscore = (float*)carve((size_t)NIMG * PROP_BEFORE * 4);
    int*      topidx   = (int*)carve((size_t)NIMG * PROP_BEFORE * 4);
    float*    topbox   = (float*)carve((size_t)NIMG * PROP_BEFORE * 4 * 4);
    unsigned* mask     = (unsigned*)carve((size_t)NIMG * PROP_BEFORE * NMS_WORDS * 4);

    dim3 b256(256);

    // weight repack into fragment order: 32*9*128*32 threads
    int rp_tot = NOCT * 9 * 128 * 32;
    hipLaunchKernelGGL(repack_w_kernel, dim3(rp_tot / 256), b256, 0, stream,
                       w1, apack);

    // 3x3 conv: 79 strips x 4 images, 274KB dynamic LDS per block
    dim3 g3((NPIX + STRIP - 1) / STRIP, 1, NIMG);
    hipLaunchKernelGGL(conv3x3_wmma_kernel, g3, b256,
                       TILE_PX * CIN * sizeof(float), stream,
                       x, apack, b1, inter);

    // 1x1 heads
    dim3 gs((157 + 7) / 8, 2, NIMG);   // score: 18 ch -> 2 oc tiles
    hipLaunchKernelGGL(conv1x1_wmma_kernel, gs, b256, 0, stream,
                       inter, w_score, b_score, out_score, 18, 0);
    dim3 gl((157 + 7) / 8, 3, NIMG);   // loc: 36 ch -> 3 oc tiles
    hipLaunchKernelGGL(conv1x1_wmma_kernel, gl, b256, 0, stream,
                       inter, w_loc, b_loc, out_loc, 36, 1);

    // decode + anchors + fg
    int tot = NIMG * R_TOT;
    hipLaunchKernelGGL(decode_kernel, dim3((tot + 255) / 256), b256, 0, stream,
                       out_loc, out_score, out_anchor, boxes, fg, img_h, img_w);

    // full bitonic sort per image in 256KB LDS
    hipLaunchKernelGGL(sort_topk_kernel, dim3(NIMG), dim3(1024),
                       SORT_N * 8, stream, fg, boxes, topscore, topidx, topbox);

    // NMS bitmask + greedy scan
    int mtot = NIMG * PROP_BEFORE * NMS_WORDS;
    hipLaunchKernelGGL(nms_mask_kernel, dim3((mtot + 255) / 256), b256, 0,
                       stream, topbox, mask);
    hipLaunchKernelGGL(nms_scan_kernel, dim3(NIMG), b256, 0, stream,
                       topscore, topbox, mask, out_rois, out_ridx);
}